// ViterbiLoss_16183436771605
// MI455X (gfx1250) — compile-verified
//
#include <hip/hip_runtime.h>
#include <cstdint>
#include <cstddef>

// Problem constants (match reference)
#define B_ 128
#define T_ 1024
#define K_ 24
#define KK_ 576          // K*K floats per (b,t) tile = 2304 bytes
#define START_TAG_ 22
#define END_TAG_ 23

#if defined(__HIP_DEVICE_COMPILE__) && defined(__gfx1250__) &&                 \
    __has_builtin(__builtin_amdgcn_tensor_load_to_lds) &&                      \
    __has_builtin(__builtin_amdgcn_s_wait_tensorcnt)
#define HAVE_TDM 1
#else
#define HAVE_TDM 0
#endif

#if defined(__HIP_DEVICE_COMPILE__) && defined(__gfx1250__) &&                 \
    __has_builtin(__builtin_amdgcn_global_load_async_to_lds_b128) &&           \
    __has_builtin(__builtin_amdgcn_global_load_async_to_lds_b64) &&            \
    __has_builtin(__builtin_amdgcn_s_wait_asynccnt)
#define HAVE_ASYNC 1
#else
#define HAVE_ASYNC 0
#endif

// Tile-ring depth (power of 2). TDM: 1 TENSORcnt slot/tile -> deep pipeline
// (16 tiles = 36.9KB LDS, covers ~0.6us+ of HBM latency at ~40ns/step).
// Async fallback: 5 ASYNCcnt slots/tile, 6-bit counter -> cap depth at 8.
#if HAVE_TDM
#define DEPTH 16
#elif HAVE_ASYNC
#define DEPTH 8
#else
#define DEPTH 2
#endif
#define IPT 5            // async instructions per tile in the ASYNC path

typedef unsigned int u32x4 __attribute__((vector_size(4 * sizeof(int))));
typedef int i32x8 __attribute__((vector_size(8 * sizeof(int))));
typedef int i32x4 __attribute__((vector_size(4 * sizeof(int))));
typedef int v4i_t __attribute__((vector_size(4 * sizeof(int))));
typedef int v2i_t __attribute__((vector_size(2 * sizeof(int))));

// Issue one 2304-byte tile (576 f32) global -> LDS.
__device__ __forceinline__ void issue_tile(const float* gsrc, float* lds, int lane) {
#if HAVE_TDM
  (void)lane;  // TDM is wave-level DMA; EXEC ignored, operands uniform (SGPRs)
  unsigned long long ga = (unsigned long long)(uintptr_t)gsrc;
  unsigned int lds_off = (unsigned int)(uintptr_t)
      (__attribute__((address_space(3))) char*)(char*)lds;  // LDS byte offset

  // D# group 0: count=1(valid user desc), lds_addr, global_addr[56:0], type=2
  u32x4 g0 = {
      1u,
      lds_off,
      (unsigned int)(ga & 0xffffffffull),
      (unsigned int)((ga >> 32) & 0x01ffffffull) | (2u << 30)};

  // D# group 1: mask=0, data_size=2(4B), dim0=tile0=576, dim1=tile1=1
  i32x8 g1 = {
      (int)0x00020000u,                       // data_size=4B, all flags 0
      (int)(576u << 16),                      // tensor_dim0[15:0] in [31:16]
      (int)(1u << 16),                        // dim0[31:16]=0 | tensor_dim1=1
      (int)(576u << 16),                      // dim1[31:16]=0 | tile_dim0=576
      1,                                      // tile_dim1=1 | tile_dim2=0
      576,                                    // tensor_dim0_stride[31:0]
      (int)(576u << 16),                      // stride0[47:32]=0 | stride1[15:0]
      0};                                     // tensor_dim1_stride[47:16]=0

  // D# group 2: tensor_dim2=1, tensor_dim3=1, dim2_stride=576, tile_dim3=0
  i32x4 g2 = {1, 1, 576, 0};
  // D# group 3: dim3_stride=576, tensor_dim4=1, tile_dim4=0
  i32x4 g3 = {576, (int)(1u << 16), 0, 0};
  // 5th operand group (VADDR4 equivalent): unused -> zeros
  i32x8 g4 = {0, 0, 0, 0, 0, 0, 0, 0};

  __builtin_amdgcn_tensor_load_to_lds(g0, g1, g2, g3, g4, 0);
#elif HAVE_ASYNC
  // 32 lanes x 16B x 4 instrs = 2048B, then 32 lanes x 8B = 256B.
  __attribute__((address_space(1))) char* gc =
      (__attribute__((address_space(1))) char*)(char*)const_cast<float*>(gsrc);
  __attribute__((address_space(3))) char* lc =
      (__attribute__((address_space(3))) char*)(char*)lds;

  __attribute__((address_space(1))) v4i_t* g =
      (__attribute__((address_space(1))) v4i_t*)(gc + lane * 16);
  __attribute__((address_space(3))) v4i_t* l =
      (__attribute__((address_space(3))) v4i_t*)(lc + lane * 16);
  __builtin_amdgcn_global_load_async_to_lds_b128(g, l, 0, 0);
  __builtin_amdgcn_global_load_async_to_lds_b128(g, l, 512, 0);
  __builtin_amdgcn_global_load_async_to_lds_b128(g, l, 1024, 0);
  __builtin_amdgcn_global_load_async_to_lds_b128(g, l, 1536, 0);

  __attribute__((address_space(1))) v2i_t* g2p =
      (__attribute__((address_space(1))) v2i_t*)(gc + 2048 + lane * 8);
  __attribute__((address_space(3))) v2i_t* l2p =
      (__attribute__((address_space(3))) v2i_t*)(lc + 2048 + lane * 8);
  __builtin_amdgcn_global_load_async_to_lds_b64(g2p, l2p, 0, 0);
#else
  #pragma unroll
  for (int k = 0; k < 18; ++k) lds[lane + 32 * k] = gsrc[lane + 32 * k];
#endif
}

// One wave32 per batch. Lane j (< 24) owns tag j of `prev`.
__global__ __launch_bounds__(32) void crf_forward_kernel(
    const float* __restrict__ scores, const int* __restrict__ lengths,
    float* __restrict__ endv) {
  const int b    = blockIdx.x;
  const int lane = threadIdx.x;
  const int len  = lengths[b];          // >= 1, uniform over the wave
  const int jj   = lane < K_ ? lane : (K_ - 1);

  __shared__ __align__(16) float tiles[DEPTH][KK_];
  const float* base = scores + (size_t)b * T_ * KK_;

  // Prime the ring (tail-clamped so exactly one tile's ops per slot in flight).
  #pragma unroll
  for (int p = 0; p < DEPTH; ++p) {
    int pt = p < len ? p : len - 1;
    issue_tile(base + (size_t)pt * KK_, tiles[p], lane);
  }

  float prev = 0.0f;
  for (int t = 0; t < len; ++t) {
#if HAVE_TDM
    // TDM ops from one wave complete in order: <= DEPTH-1 outstanding
    // implies tiles 0..t are resident.
    __builtin_amdgcn_s_wait_tensorcnt(DEPTH - 1);
#elif HAVE_ASYNC
    __builtin_amdgcn_s_wait_asynccnt((DEPTH - 1) * IPT);
#else
    __syncthreads();
#endif
    const float* tile = tiles[t & (DEPTH - 1)];

    if (t == 0) {
      prev = tile[START_TAG_ * K_ + jj];          // prev_j = scores[b,0,START,j]
    } else {
      // m = max over lanes 0..23 of prev (stability shift)
      float m = (lane < K_) ? prev : -3.0e38f;
      #pragma unroll
      for (int off = 16; off > 0; off >>= 1)
        m = fmaxf(m, __shfl_xor(m, off, 32));
      const float pm = prev - m;

      // acc_j = sum_i exp(prev_i - m + s[i][j]); 4 accumulators break the chain.
      float a0 = 0.f, a1 = 0.f, a2 = 0.f, a3 = 0.f;
      #pragma unroll
      for (int i = 0; i < K_; i += 4) {
        a0 += __expf(__shfl(pm, i + 0, 32) + tile[(i + 0) * K_ + jj]);
        a1 += __expf(__shfl(pm, i + 1, 32) + tile[(i + 1) * K_ + jj]);
        a2 += __expf(__shfl(pm, i + 2, 32) + tile[(i + 2) * K_ + jj]);
        a3 += __expf(__shfl(pm, i + 3, 32) + tile[(i + 3) * K_ + jj]);
      }
      const float nv = m + __logf((a0 + a1) + (a2 + a3));
      prev = (lane < K_) ? nv : prev;
    }

#if !HAVE_TDM && !HAVE_ASYNC
    __syncthreads();   // fallback: protect WAR on the ring slot
#endif
    // Keep exactly DEPTH tiles in flight (clamp duplicates near the tail so
    // the constant-immediate counter wait above stays correct).
    int nt = t + DEPTH; if (nt > len - 1) nt = len - 1;
    issue_tile(base + (size_t)nt * KK_, tiles[(t + DEPTH) & (DEPTH - 1)], lane);
  }

  if (lane == END_TAG_) endv[b] = prev;
}

// Gold score: one thread per (b,t); deterministic block tree-reduce.
__global__ __launch_bounds__(256) void gold_kernel(
    const float* __restrict__ scores, const int* __restrict__ targets,
    const int* __restrict__ lengths, float* __restrict__ partial) {
  __shared__ float red[256];
  const int idx = blockIdx.x * 256 + threadIdx.x;   // grid covers exactly B*T
  const int b = idx >> 10;                          // T_ == 1024
  const int t = idx & (T_ - 1);
  float v = 0.0f;
  if (t < lengths[b]) v = scores[(size_t)idx * KK_ + targets[idx]];
  red[threadIdx.x] = v;
  __syncthreads();
  #pragma unroll
  for (int s = 128; s > 0; s >>= 1) {
    if (threadIdx.x < s) red[threadIdx.x] += red[threadIdx.x + s];
    __syncthreads();
  }
  if (threadIdx.x == 0) partial[blockIdx.x] = red[0];
}

// out = (sum_b end[b] - sum gold_partials) / B   (fixed-order => deterministic)
__global__ __launch_bounds__(256) void finalize_kernel(
    const float* __restrict__ endv, const float* __restrict__ partial,
    float* __restrict__ out) {
  __shared__ float red[256];
  float acc = 0.0f;
  for (int i = threadIdx.x; i < (B_ * T_) / 256; i += 256) acc -= partial[i];
  if (threadIdx.x < B_) acc += endv[threadIdx.x];
  red[threadIdx.x] = acc;
  __syncthreads();
  #pragma unroll
  for (int s = 128; s > 0; s >>= 1) {
    if (threadIdx.x < s) red[threadIdx.x] += red[threadIdx.x + s];
    __syncthreads();
  }
  if (threadIdx.x == 0) out[0] = red[0] * (1.0f / (float)B_);
}

extern "C" void kernel_launch(void* const* d_in, const int* in_sizes, int n_in,
                              void* d_out, int out_size, void* d_ws, size_t ws_size,
                              hipStream_t stream) {
  (void)in_sizes; (void)n_in; (void)out_size; (void)ws_size;
  const float* scores  = (const float*)d_in[0];
  const int*   targets = (const int*)d_in[1];
  const int*   lengths = (const int*)d_in[2];
  float* out = (float*)d_out;
  float* ws  = (float*)d_ws;
  float* endv    = ws;        // 128 floats
  float* partial = ws + B_;   // 512 floats

  hipLaunchKernelGGL(crf_forward_kernel, dim3(B_), dim3(32), 0, stream,
                     scores, lengths, endv);
  hipLaunchKernelGGL(gold_kernel, dim3((B_ * T_) / 256), dim3(256), 0, stream,
                     scores, targets, lengths, partial);
  hipLaunchKernelGGL(finalize_kernel, dim3(1), dim3(256), 0, stream,
                     endv, partial, out);
}